// SeparationLoss_3642132267019
// MI455X (gfx1250) — compile-verified
//
#include <hip/hip_runtime.h>
#include <hip/hip_bf16.h>

typedef __attribute__((ext_vector_type(2))) float v2f;
typedef __attribute__((ext_vector_type(8))) float v8f;

#define N_ROWS 4096
#define D_DIM  2048
#define K_DIM  64
#define NTILES (N_ROWS / 16)        // 256 row tiles
#define KCHUNK (D_DIM / 8)          // 256: per-wave K range in z_kernel

// ---------------------------------------------------------------------------
// Kernel 1: A = y^T y  (64x64), also zero the output accumulator.
// y is (2048, 64) row-major. Block i computes row i of A; thread j -> coalesced.
// ---------------------------------------------------------------------------
__global__ void yty_kernel(const float* __restrict__ y, float* __restrict__ A,
                           float* __restrict__ out) {
    const int i = blockIdx.x;   // 0..63
    const int j = threadIdx.x;  // 0..63
    float acc = 0.0f;
    for (int r = 0; r < D_DIM; ++r) {
        acc = fmaf(y[r * K_DIM + i], y[r * K_DIM + j], acc);
    }
    A[i * K_DIM + j] = acc;
    if (i == 0 && j == 0) out[0] = 0.0f;
}

// ---------------------------------------------------------------------------
// Kernel 2: Gauss-Jordan inversion of 64x64 SPD matrix in LDS, one workgroup.
// ---------------------------------------------------------------------------
__global__ void invert64_kernel(const float* __restrict__ A,
                                float* __restrict__ Ainv) {
    __shared__ float M[64][129];   // augmented [A | I], padded stride
    __shared__ float fac[64];
    __shared__ float pivinv;
    const int tid = threadIdx.x;   // 256 threads

    for (int e = tid; e < 64 * 64; e += 256) {
        const int r = e >> 6, c = e & 63;
        M[r][c]      = A[e];
        M[r][64 + c] = (r == c) ? 1.0f : 0.0f;
    }
    __syncthreads();

    for (int p = 0; p < 64; ++p) {
        if (tid == 0) pivinv = 1.0f / M[p][p];
        __syncthreads();
        if (tid < 128) M[p][tid] *= pivinv;           // scale pivot row
        __syncthreads();
        if (tid < 64) fac[tid] = M[tid][p];           // save column factors
        __syncthreads();
        for (int e = tid; e < 64 * 128; e += 256) {   // eliminate
            const int r = e >> 7, c = e & 127;
            if (r != p) M[r][c] = fmaf(-fac[r], M[p][c], M[r][c]);
        }
        __syncthreads();
    }

    for (int e = tid; e < 64 * 64; e += 256) {
        const int r = e >> 6, c = e & 63;
        Ainv[e] = M[r][64 + c];
    }
}

// ---------------------------------------------------------------------------
// Kernel 3: z = x @ y  (4096x64), fp32 WMMA 16x16x4, K-split for occupancy.
// Block rt owns row tile rt. Its 8 waves each accumulate a partial 16x64 tile
// over a disjoint 256-wide K chunk (x read exactly once, fully pipelined),
// write partials to LDS, then the block reduces 8 partials -> z tile.
// 256 blocks x 8 waves = 2048 waves in flight to saturate 23.3 TB/s.
// ---------------------------------------------------------------------------
__global__ void __launch_bounds__(256)
z_kernel(const float* __restrict__ x, const float* __restrict__ y,
         float* __restrict__ z) {
    __shared__ float part[8][16 * 64];   // 32 KB of 320 KB LDS
    const int waveid = threadIdx.x >> 5; // 0..7 -> K chunk
    const int lane   = threadIdx.x & 31;
    const int rt     = blockIdx.x;       // 0..255 row tile
    const int m  = lane & 15;            // A row / B col within tile
    const int kb = (lane >> 4) << 1;     // K sub-offset: 0 (lanes 0-15) / 2 (16-31)
    const int row    = rt * 16 + m;
    const int kstart = waveid * KCHUNK;

    v8f acc0 = {}, acc1 = {}, acc2 = {}, acc3 = {};
    const float* xr = x + (size_t)row * D_DIM;

    for (int k0 = kstart; k0 < kstart + KCHUNK; k0 += 4) {
        __builtin_prefetch(xr + k0 + 128, 0, 0);   // global_prefetch_b8
        v2f a;
        a.x = xr[k0 + kb];
        a.y = xr[k0 + kb + 1];
        const float* yb0 = y + (size_t)(k0 + kb)     * K_DIM;
        const float* yb1 = y + (size_t)(k0 + kb + 1) * K_DIM;
        v2f b0, b1, b2, b3;
        b0.x = yb0[m];       b0.y = yb1[m];
        b1.x = yb0[16 + m];  b1.y = yb1[16 + m];
        b2.x = yb0[32 + m];  b2.y = yb1[32 + m];
        b3.x = yb0[48 + m];  b3.y = yb1[48 + m];
        acc0 = __builtin_amdgcn_wmma_f32_16x16x4_f32(false, a, false, b0, (short)0, acc0, false, false);
        acc1 = __builtin_amdgcn_wmma_f32_16x16x4_f32(false, a, false, b1, (short)0, acc1, false, false);
        acc2 = __builtin_amdgcn_wmma_f32_16x16x4_f32(false, a, false, b2, (short)0, acc2, false, false);
        acc3 = __builtin_amdgcn_wmma_f32_16x16x4_f32(false, a, false, b3, (short)0, acc3, false, false);
    }

    // C/D layout: VGPR v, lane L -> row-in-tile v + 8*(L>=16), col L%16
    const int rloc = (lane >> 4) << 3;
    const int c    = lane & 15;
    float* pw = &part[waveid][0];
    for (int v = 0; v < 8; ++v) {
        float* pr = pw + (rloc + v) * 64;
        pr[c]      = acc0[v];
        pr[16 + c] = acc1[v];
        pr[32 + c] = acc2[v];
        pr[48 + c] = acc3[v];
    }
    __syncthreads();

    // cross-wave reduction: 1024 elements, 4 per thread (deterministic order)
    for (int e = threadIdx.x; e < 16 * 64; e += 256) {
        float s = part[0][e] + part[1][e] + part[2][e] + part[3][e]
                + part[4][e] + part[5][e] + part[6][e] + part[7][e];
        z[(size_t)(rt * 16 + (e >> 6)) * K_DIM + (e & 63)] = s;
    }
}

// ---------------------------------------------------------------------------
// Kernel 4: u = z @ Ainv  (4096x64 @ 64x64), fp32 WMMA, K = 64.
// One wave per 16x16 output tile -> 1024 waves (z/Ainv are L2-hot).
// ---------------------------------------------------------------------------
__global__ void __launch_bounds__(256)
u_kernel(const float* __restrict__ z, const float* __restrict__ Ainv,
         float* __restrict__ u) {
    const int wave = (blockIdx.x * blockDim.x + threadIdx.x) >> 5;  // 0..1023
    const int lane = threadIdx.x & 31;
    const int rt = wave >> 2;        // row tile 0..255
    const int ct = wave & 3;         // col tile 0..3
    const int m  = lane & 15;
    const int kb = (lane >> 4) << 1;

    v8f acc = {};
    const float* zr = z + (size_t)(rt * 16 + m) * K_DIM;

    for (int k0 = 0; k0 < K_DIM; k0 += 4) {
        v2f a, b;
        a.x = zr[k0 + kb];
        a.y = zr[k0 + kb + 1];
        b.x = Ainv[(size_t)(k0 + kb)     * K_DIM + ct * 16 + m];
        b.y = Ainv[(size_t)(k0 + kb + 1) * K_DIM + ct * 16 + m];
        acc = __builtin_amdgcn_wmma_f32_16x16x4_f32(false, a, false, b, (short)0, acc, false, false);
    }

    const int rbase = rt * 16 + ((lane >> 4) << 3);
    const int c     = ct * 16 + (lane & 15);
    for (int v = 0; v < 8; ++v) {
        u[(size_t)(rbase + v) * K_DIM + c] = acc[v];
    }
}

// ---------------------------------------------------------------------------
// Kernel 5: sq_norms[i] = dot(u_i, z_i)  (== gram[i,i])
// ---------------------------------------------------------------------------
__global__ void sqn_kernel(const float* __restrict__ u,
                           const float* __restrict__ z,
                           float* __restrict__ sqn) {
    const int i = blockIdx.x * blockDim.x + threadIdx.x;  // 0..4095
    const float* ur = u + (size_t)i * K_DIM;
    const float* zr = z + (size_t)i * K_DIM;
    float acc = 0.0f;
    for (int c = 0; c < K_DIM; ++c) acc = fmaf(ur[c], zr[c], acc);
    sqn[i] = acc;
}

// ---------------------------------------------------------------------------
// Kernel 6: fused gram-tile + distance + signed masked reduction.
// Triangular-compacted: P = 256*257/2 = 32896 wave-tiles = 4112 blocks x 8
// waves, no wasted lower-triangle waves. Wave w recovers (ti, tj>=ti) from a
// wave-uniform inverse-triangular map (EXEC stays all-ones around WMMA).
// gram tile = u_tile @ z_tile^T via 16 fp32 WMMAs (K=64), fused
// sq -> sqrt -> sign -> mask, wave32 shfl reduction, atomicAdd to d_out.
// ---------------------------------------------------------------------------
__device__ __forceinline__ int tri_base(int ti) {
    // offset of first pair with row tile ti (pairs ordered ti asc, tj asc)
    return ti * NTILES - ((ti * (ti - 1)) >> 1);
}

__global__ void __launch_bounds__(256)
loss_kernel(const float* __restrict__ u, const float* __restrict__ z,
            const float* __restrict__ sqn, const int* __restrict__ t,
            float* __restrict__ out) {
    const int waveid = threadIdx.x >> 5;
    const int lane   = threadIdx.x & 31;
    const int w = blockIdx.x * 8 + waveid;    // 0..32895, wave-uniform

    // invert triangular index: largest ti with tri_base(ti) <= w
    const float Tf = (float)NTILES + 0.5f;
    int ti = (int)(Tf - sqrtf(Tf * Tf - 2.0f * (float)w));
    if (ti < 0) ti = 0;
    if (ti > NTILES - 1) ti = NTILES - 1;
    while (ti > 0 && tri_base(ti) > w) --ti;
    while (ti < NTILES - 1 && tri_base(ti + 1) <= w) ++ti;
    const int tj = ti + (w - tri_base(ti));   // tj >= ti

    const int m  = lane & 15;
    const int kb = (lane >> 4) << 1;
    const float* ua = u + (size_t)(ti * 16 + m) * K_DIM;  // A: u rows
    const float* zb = z + (size_t)(tj * 16 + m) * K_DIM;  // B = z^T: cols are z rows

    v8f acc = {};
    for (int k0 = 0; k0 < K_DIM; k0 += 4) {
        v2f a, b;
        a.x = ua[k0 + kb];
        a.y = ua[k0 + kb + 1];
        b.x = zb[k0 + kb];
        b.y = zb[k0 + kb + 1];
        acc = __builtin_amdgcn_wmma_f32_16x16x4_f32(false, a, false, b, (short)0, acc, false, false);
    }

    const int nn = lane & 15;
    const int j  = tj * 16 + nn;
    const float sqj = sqn[j];
    const int   tjv = t[j];
    const int   rb  = ti * 16 + ((lane >> 4) << 3);

    float sum = 0.0f;
    for (int v = 0; v < 8; ++v) {
        const int i = rb + v;
        if (j > i) {                           // strict upper triangle
            float sq = sqn[i] + sqj - 2.0f * acc[v];
            sq = fmaxf(sq, 0.0f);
            const float d = sqrtf(sq);
            sum += (t[i] == tjv) ? d : -d;
        }
    }

    // wave32 reduction
    for (int off = 16; off > 0; off >>= 1)
        sum += __shfl_xor(sum, off, 32);
    if (lane == 0) atomicAdd(out, sum);
}

// ---------------------------------------------------------------------------
extern "C" void kernel_launch(void* const* d_in, const int* in_sizes, int n_in,
                              void* d_out, int out_size, void* d_ws, size_t ws_size,
                              hipStream_t stream) {
    const float* x = (const float*)d_in[0];   // (4096, 2048)
    const int*   t = (const int*)d_in[1];     // (4096,)
    const float* y = (const float*)d_in[2];   // (2048, 64)
    float* out = (float*)d_out;               // scalar loss

    float* ws   = (float*)d_ws;
    float* A    = ws;                          // 64*64
    float* Ainv = A + 64 * 64;                 // 64*64
    float* z    = Ainv + 64 * 64;              // 4096*64
    float* u    = z + (size_t)N_ROWS * K_DIM;  // 4096*64
    float* sqn  = u + (size_t)N_ROWS * K_DIM;  // 4096

    yty_kernel     <<<64, 64, 0, stream>>>(y, A, out);
    invert64_kernel<<<1, 256, 0, stream>>>(A, Ainv);
    z_kernel       <<<NTILES, 256, 0, stream>>>(x, y, z);          // 2048 waves
    u_kernel       <<<128, 256, 0, stream>>>(z, Ainv, u);          // 1024 waves
    sqn_kernel     <<<16, 256, 0, stream>>>(u, z, sqn);
    const int P = NTILES * (NTILES + 1) / 2;                       // 32896
    loss_kernel    <<<P / 8, 256, 0, stream>>>(u, z, sqn, t, out); // 4112 blocks
}